// GraphConvolutionalNetwork_69887707840880
// MI455X (gfx1250) — compile-verified
//
#include <hip/hip_runtime.h>

#define N_NODES 100000
#define N_EDGES 1600000
#define D       128
#define N_GRAPHS 256

typedef __attribute__((ext_vector_type(2))) float v2f;
typedef __attribute__((ext_vector_type(8))) float v8f;

// ---------------------------------------------------------------------------
// Inline-asm CDNA5 atomics (guarantee native L2 float atomics, device scope)
// ---------------------------------------------------------------------------
__device__ __forceinline__ void atom_add_f32(float* p, float v) {
    asm volatile("global_atomic_add_f32 %0, %1, off scope:SCOPE_DEV"
                 :: "v"(p), "v"(v) : "memory");
}
__device__ __forceinline__ void atom_max_f32(float* p, float v) {
    asm volatile("global_atomic_max_num_f32 %0, %1, off scope:SCOPE_DEV"
                 :: "v"(p), "v"(v) : "memory");
}

// ---------------------------------------------------------------------------
// Utility kernels
// ---------------------------------------------------------------------------
__global__ void fill_f32_kernel(float* __restrict__ p, float val, int n4) {
    int i = blockIdx.x * blockDim.x + threadIdx.x;
    if (i < n4) {
        float4 v = make_float4(val, val, val, val);
        *(float4*)(p + 4 * (size_t)i) = v;
    }
}

// ---------------------------------------------------------------------------
// Edge scatter-add: agg[dst] += feat[src].  One wave (32 lanes) per edge,
// each lane moves a float4 (32*4 = 128 = D).  feat & agg are L2-resident.
// ---------------------------------------------------------------------------
__global__ __launch_bounds__(256)
void scatter_add_kernel(const float* __restrict__ feat,
                        const long long* __restrict__ edge_index,
                        float* __restrict__ agg) {
    const int tid  = blockIdx.x * blockDim.x + threadIdx.x;
    const int e    = tid >> 5;
    const int lane = tid & 31;
    if (e >= N_EDGES) return;
    const long long s = edge_index[e];            // src row
    const long long d = edge_index[N_EDGES + e];  // dst row
    const float4 v = *(const float4*)(feat + s * D + lane * 4);
    float* o = agg + d * D + lane * 4;
    atom_add_f32(o + 0, v.x);
    atom_add_f32(o + 1, v.y);
    atom_add_f32(o + 2, v.z);
    atom_add_f32(o + 3, v.w);
}

// ---------------------------------------------------------------------------
// Fused dual-GEMM:  C = A1 @ W1^T + A2 @ W2^T + bias  (all f32, full precision)
// using V_WMMA_F32_16X16X4_F32.  One wave computes one 16x16 output tile;
// block = 8 waves = one 16-row x 128-col strip; grid.x = 100000/16 = 6250.
//
// Fragment layouts (ISA 7.12.2):
//   A 16x4 : lane&15 = M row; lane>=16 selects K={2,3}; v2f = {K, K+1}
//   B 4x16 : lane&15 = N col; lane>=16 selects K={2,3}; B[k,n] = W[n,k]
//            -> contiguous float2 from row-major W[n*D + k]
//   C 16x16: acc[v] is row (v + 8*(lane>=16)), col = lane&15
//
// EPILOGUE 0: relu(C) -> out_h          (layer 1)
// EPILOGUE 1: atomic max into out_pool[batch[row]]   (layer 2 + global_max_pool)
// ---------------------------------------------------------------------------
template <int EPILOGUE>
__global__ __launch_bounds__(256)
void gemm_dual_kernel(const float* __restrict__ A1, const float* __restrict__ W1,
                      const float* __restrict__ A2, const float* __restrict__ W2,
                      const float* __restrict__ bias,
                      float* __restrict__ out_h,
                      const long long* __restrict__ batch,
                      float* __restrict__ out_pool) {
    const int wave  = threadIdx.x >> 5;   // 0..7 -> column tile
    const int lane  = threadIdx.x & 31;
    const int l15   = lane & 15;
    const int lhalf = lane >> 4;          // 0: K pair {0,1}, 1: K pair {2,3}
    const int row0  = blockIdx.x * 16;
    const int col0  = wave * 16;

    const float* a1p = A1 + (size_t)(row0 + l15) * D + lhalf * 2;
    const float* a2p = A2 + (size_t)(row0 + l15) * D + lhalf * 2;
    const float* w1p = W1 + (size_t)(col0 + l15) * D + lhalf * 2;  // B1[k,n]=W1[n,k]
    const float* w2p = W2 + (size_t)(col0 + l15) * D + lhalf * 2;  // B2[k,n]=W2[n,k]

    v8f acc = {};
#pragma unroll
    for (int k = 0; k < D; k += 4) {
        v2f a1 = *(const v2f*)(a1p + k);
        v2f b1 = *(const v2f*)(w1p + k);
        acc = __builtin_amdgcn_wmma_f32_16x16x4_f32(
                  false, a1, false, b1, (short)0, acc, false, false);
        v2f a2 = *(const v2f*)(a2p + k);
        v2f b2 = *(const v2f*)(w2p + k);
        acc = __builtin_amdgcn_wmma_f32_16x16x4_f32(
                  false, a2, false, b2, (short)0, acc, false, false);
    }

    const float b = bias[col0 + l15];

    if (EPILOGUE == 0) {
#pragma unroll
        for (int v = 0; v < 8; ++v) {
            const int r = row0 + v + lhalf * 8;
            float val = acc[v] + b;
            val = fmaxf(val, 0.0f);                       // ReLU
            out_h[(size_t)r * D + col0 + l15] = val;
        }
    } else {
#pragma unroll
        for (int v = 0; v < 8; ++v) {
            const int r = row0 + v + lhalf * 8;
            const float val = acc[v] + b;
            const long long g = batch[r];                 // sorted segment id
            atom_max_f32(out_pool + (size_t)g * D + col0 + l15, val);
        }
    }
}

// ---------------------------------------------------------------------------
// Host launch
// ---------------------------------------------------------------------------
extern "C" void kernel_launch(void* const* d_in, const int* in_sizes, int n_in,
                              void* d_out, int out_size, void* d_ws, size_t ws_size,
                              hipStream_t stream) {
    const float*     x       = (const float*)d_in[0];
    const long long* ei      = (const long long*)d_in[1];
    const long long* batch   = (const long long*)d_in[2];
    const float*     W_rel1  = (const float*)d_in[3];
    const float*     b_rel1  = (const float*)d_in[4];
    const float*     W_root1 = (const float*)d_in[5];
    const float*     W_rel2  = (const float*)d_in[6];
    const float*     b_rel2  = (const float*)d_in[7];
    const float*     W_root2 = (const float*)d_in[8];
    float* out = (float*)d_out;

    float* agg = (float*)d_ws;                     // N_NODES * D floats
    float* h   = agg + (size_t)N_NODES * D;        // N_NODES * D floats

    const int feat4    = (N_NODES * D) / 4;        // 3.2M float4 slots
    const int featBlks = feat4 / 256;              // 12500
    const int edgeBlks = (N_EDGES * 32) / 256;     // 200000
    const int gemmBlks = N_NODES / 16;             // 6250
    const int out4     = (N_GRAPHS * D) / 4;       // 8192
    const int outBlks  = out4 / 256;               // 32

    // ---- Layer 1 ----
    fill_f32_kernel<<<featBlks, 256, 0, stream>>>(agg, 0.0f, feat4);
    scatter_add_kernel<<<edgeBlks, 256, 0, stream>>>(x, ei, agg);
    gemm_dual_kernel<0><<<gemmBlks, 256, 0, stream>>>(
        agg, W_rel1, x, W_root1, b_rel1, h, nullptr, nullptr);

    // ---- Layer 2 (agg buffer reused; stream order serializes) ----
    fill_f32_kernel<<<featBlks, 256, 0, stream>>>(agg, 0.0f, feat4);
    scatter_add_kernel<<<edgeBlks, 256, 0, stream>>>(h, ei, agg);
    fill_f32_kernel<<<outBlks, 256, 0, stream>>>(out, -INFINITY, out4);
    gemm_dual_kernel<1><<<gemmBlks, 256, 0, stream>>>(
        agg, W_rel2, h, W_root2, b_rel2, nullptr, batch, out);
}